// NodeBlock_45509473468802
// MI455X (gfx1250) — compile-verified
//
#include <hip/hip_runtime.h>

typedef float v2f __attribute__((ext_vector_type(2)));
typedef float v4f __attribute__((ext_vector_type(4)));
typedef float v8f __attribute__((ext_vector_type(8)));

#define N_NODES 50000
#define N_EDGES 800000
#define D_NODE  64
#define D_EDGE  64
#define D_IN    128   // D_NODE + D_EDGE
#define HIDDEN  128

// ---------------------------------------------------------------------------
// Kernel 1: zero the agg scratch buffer (d_ws is poisoned, we need zeros).
// ---------------------------------------------------------------------------
__global__ void zero_f32_kernel(float* __restrict__ p, int n) {
    int i = blockIdx.x * blockDim.x + threadIdx.x;
    int stride = gridDim.x * blockDim.x;
    for (; i < n; i += stride) p[i] = 0.0f;
}

// ---------------------------------------------------------------------------
// Kernel 2: scatter-add edge features into agg by BOTH src and dst index.
// One thread per (edge, 4-float chunk): coalesced 16B edge reads, then
// 8 native global_atomic_add_f32 (agg is 12.8MB -> lives in L2).
// ---------------------------------------------------------------------------
__global__ void scatter_add_kernel(const float* __restrict__ x_edge,
                                   const long long* __restrict__ eidx,
                                   float* __restrict__ agg) {
    int tid = blockIdx.x * blockDim.x + threadIdx.x;
    if (tid >= N_EDGES * (D_EDGE / 4)) return;
    int e = tid >> 4;              // edge id      (D_EDGE/4 == 16 chunks)
    int c = (tid & 15) << 2;       // feature base (0,4,...,60)

    v4f v = *(const v4f*)(x_edge + (size_t)e * D_EDGE + c);
    int s = (int)eidx[e];                 // edge_index[0][e]
    int d = (int)eidx[N_EDGES + e];       // edge_index[1][e]

    float* ps = agg + (size_t)s * D_EDGE + c;
    float* pd = agg + (size_t)d * D_EDGE + c;
    unsafeAtomicAdd(ps + 0, v.x);
    unsafeAtomicAdd(ps + 1, v.y);
    unsafeAtomicAdd(ps + 2, v.z);
    unsafeAtomicAdd(ps + 3, v.w);
    unsafeAtomicAdd(pd + 0, v.x);
    unsafeAtomicAdd(pd + 1, v.y);
    unsafeAtomicAdd(pd + 2, v.z);
    unsafeAtomicAdd(pd + 3, v.w);
}

// ---------------------------------------------------------------------------
// Kernel 3: fused MLP  out = relu(concat(x_node,agg) @ W1 + b1) @ W2 + b2
// using V_WMMA_F32_16X16X4_F32. One wave per 16-row tile, 4 waves/block.
//
// A-matrix (16x4 f32) per-lane layout (ISA 7.12.2):
//   lane l<16 : A[l][k0+0], A[l][k0+1]   lane l>=16 : A[l-16][k0+2], A[l-16][k0+3]
// B-matrix (4x16 f32): VGPR0 = { B[k0+0][n] | lanes0-15, B[k0+2][n] | lanes16-31 },
//                      VGPR1 = rows k0+1 / k0+3  (symmetric with A).
// C/D (16x16 f32): VGPR g = rows {g, g+8} at col n0+(lane&15).
// ---------------------------------------------------------------------------
__global__ void __launch_bounds__(128)
mlp_wmma_kernel(const float* __restrict__ x_node,
                const float* __restrict__ agg,
                const float* __restrict__ W1, const float* __restrict__ b1,
                const float* __restrict__ W2, const float* __restrict__ b2,
                float* __restrict__ out) {
    __shared__ float sH[4 * 16 * HIDDEN];   // per-wave 16x128 hidden tile

    const int lane = threadIdx.x & 31;
    const int wv   = threadIdx.x >> 5;      // wave id in block (0..3)
    const int half = lane >> 4;             // 0: lanes 0-15, 1: lanes 16-31
    const int l16  = lane & 15;

    const int rb = blockIdx.x * 64 + wv * 16;   // first row of this wave's tile
    int r = rb + l16;                           // A-matrix row for this lane
    if (r >= N_NODES) r = N_NODES - 1;          // clamp loads; stores guarded

    // ---- load A fragments for layer 1: concat(x_node, agg) row r ----------
    v2f A[32];
#pragma unroll
    for (int k = 0; k < 32; ++k) {
        int k0 = 4 * k + 2 * half;              // even -> never straddles the
        const float* src = (k0 < D_NODE)        // x_node|agg boundary at 64
            ? (x_node + (size_t)r * D_NODE + k0)
            : (agg    + (size_t)r * D_EDGE + (k0 - D_NODE));
        A[k] = *(const v2f*)src;
    }

    float* hrow = sH + wv * (16 * HIDDEN);

    // ---- layer 1: h = relu(in @ W1 + b1), 8 N-tiles of 16 -----------------
    for (int t = 0; t < 8; ++t) {
        const int n0 = 16 * t;
        v8f acc = {};
#pragma unroll
        for (int k = 0; k < 32; ++k) {
            const int kr = 4 * k + 2 * half;
            v2f b;
            b.x = W1[(kr + 0) * HIDDEN + n0 + l16];
            b.y = W1[(kr + 1) * HIDDEN + n0 + l16];
            acc = __builtin_amdgcn_wmma_f32_16x16x4_f32(
                false, A[k], false, b, (short)0, acc, false, false);
        }
        const float bias = b1[n0 + l16];
#pragma unroll
        for (int g = 0; g < 8; ++g) {
            float v = acc[g] + bias;
            v = v > 0.0f ? v : 0.0f;                      // relu
            hrow[(g + 8 * half) * HIDDEN + n0 + l16] = v; // D-layout -> LDS
        }
    }

    __syncthreads();   // order LDS writes before A-layout reads

    // ---- reload h in A-matrix layout --------------------------------------
#pragma unroll
    for (int k = 0; k < 32; ++k) {
        A[k] = *(const v2f*)(hrow + l16 * HIDDEN + 4 * k + 2 * half);
    }

    // ---- layer 2: out = h @ W2 + b2, 4 N-tiles of 16 ----------------------
    for (int t = 0; t < 4; ++t) {
        const int n0 = 16 * t;
        v8f acc = {};
#pragma unroll
        for (int k = 0; k < 32; ++k) {
            const int kr = 4 * k + 2 * half;
            v2f b;
            b.x = W2[(kr + 0) * D_NODE + n0 + l16];
            b.y = W2[(kr + 1) * D_NODE + n0 + l16];
            acc = __builtin_amdgcn_wmma_f32_16x16x4_f32(
                false, A[k], false, b, (short)0, acc, false, false);
        }
        const float bias = b2[n0 + l16];
#pragma unroll
        for (int g = 0; g < 8; ++g) {
            const int row = rb + g + 8 * half;
            if (row < N_NODES)
                out[(size_t)row * D_NODE + n0 + l16] = acc[g] + bias;
        }
    }
}

// ---------------------------------------------------------------------------
extern "C" void kernel_launch(void* const* d_in, const int* in_sizes, int n_in,
                              void* d_out, int out_size, void* d_ws, size_t ws_size,
                              hipStream_t stream) {
    const float*     x_node = (const float*)d_in[0];
    const float*     x_edge = (const float*)d_in[1];
    const long long* eidx   = (const long long*)d_in[2];  // int64 per reference
    const float*     W1     = (const float*)d_in[3];
    const float*     b1     = (const float*)d_in[4];
    const float*     W2     = (const float*)d_in[5];
    const float*     b2     = (const float*)d_in[6];
    float*           outp   = (float*)d_out;
    float*           agg    = (float*)d_ws;               // 50000*64 f32 = 12.8 MB

    (void)in_sizes; (void)n_in; (void)out_size; (void)ws_size;

    const int aggN = N_NODES * D_EDGE;
    zero_f32_kernel<<<(aggN + 255) / 256, 256, 0, stream>>>(agg, aggN);

    const int sthreads = N_EDGES * (D_EDGE / 4);
    scatter_add_kernel<<<(sthreads + 255) / 256, 256, 0, stream>>>(x_edge, eidx, agg);

    mlp_wmma_kernel<<<(N_NODES + 63) / 64, 128, 0, stream>>>(
        x_node, agg, W1, b1, W2, b2, outp);
}